// LatticeMaskExpert_85547158601817
// MI455X (gfx1250) — compile-verified
//
#include <hip/hip_runtime.h>
#include <math.h>

typedef float v2f __attribute__((ext_vector_type(2)));
typedef float v8f __attribute__((ext_vector_type(8)));

#define KDIM   25600
#define BATCH  2048
#define ACOLS  32            // 26 alphas padded to 32
#define KSPLIT 16
#define KRANGE (KDIM / KSPLIT)   // 1600
#define CK     160               // K staged in LDS per chunk
#define WAVE_K 40                // K per wave per chunk (10 WMMA steps)

// ---------------------------------------------------------------------------
// Kernel 0: alphas = bias (padded cols = 0). GEMM partials are atomically added.
__global__ void init_alphas_kernel(const float* __restrict__ b,
                                   float* __restrict__ alphas) {
    int idx = blockIdx.x * blockDim.x + threadIdx.x;
    if (idx < BATCH * ACOLS) {
        int c = idx & (ACOLS - 1);
        alphas[idx] = (c < 26) ? b[c] : 0.0f;
    }
}

// ---------------------------------------------------------------------------
// Kernel 1: alphas += x @ W via V_WMMA_F32_16X16X4_F32.
// grid = (BATCH/16 row tiles, KSPLIT k-splits), block = 128 (4 waves).
// Each wave owns a 16x32 f32 accumulator (cols padded 26->32).
__global__ void gemm_alphas_kernel(const float* __restrict__ x,
                                   const float* __restrict__ W,
                                   float* __restrict__ alphas) {
    __shared__ float xs[16][CK + 1];   // pad to break bank conflicts
    __shared__ float wsm[CK][33];      // 32 cols padded to 33

    const int tid  = threadIdx.x;
    const int lane = tid & 31;
    const int wave = tid >> 5;
    const int half = lane >> 4;        // 0: K=0,1  1: K=2,3 (A/B lane split)
    const int r    = lane & 15;        // M row (A) / N col (B)

    const int rowBase = blockIdx.x * 16;
    const int kBase   = blockIdx.y * KRANGE;

    v8f acc0 = {};   // cols 0..15
    v8f acc1 = {};   // cols 16..31

    for (int chunk = 0; chunk < KRANGE / CK; ++chunk) {
        const int k0 = kBase + chunk * CK;
        // stage x tile: 16 rows x CK k-values (coalesced in k)
        for (int l = tid; l < 16 * CK; l += 128) {
            int rr = l / CK, cc = l % CK;
            xs[rr][cc] = x[(size_t)(rowBase + rr) * KDIM + (k0 + cc)];
        }
        // stage W tile: CK x 26 -> CK x 32 (zero pad)
        for (int l = tid; l < CK * 32; l += 128) {
            int kk = l >> 5, cc = l & 31;
            wsm[kk][cc] = (cc < 26) ? W[(size_t)(k0 + kk) * 26 + cc] : 0.0f;
        }
        __syncthreads();

        const int kw = wave * WAVE_K;
        #pragma unroll
        for (int s = 0; s < WAVE_K / 4; ++s) {
            const int kb = kw + s * 4 + 2 * half;
            // A 16x4 f32: lanes 0-15 rows M, VGPR0/1 = K0/K1; lanes 16-31 = K2/K3
            v2f a;  a.x  = xs[r][kb];          a.y  = xs[r][kb + 1];
            // B 4x16 f32 (mirror layout): VGPR0 = K0|K2 row, VGPR1 = K1|K3 row
            v2f b0; b0.x = wsm[kb][r];         b0.y = wsm[kb + 1][r];
            v2f b1; b1.x = wsm[kb][16 + r];    b1.y = wsm[kb + 1][16 + r];
            acc0 = __builtin_amdgcn_wmma_f32_16x16x4_f32(false, a, false, b0,
                                                         (short)0, acc0, false, false);
            acc1 = __builtin_amdgcn_wmma_f32_16x16x4_f32(false, a, false, b1,
                                                         (short)0, acc1, false, false);
        }
        __syncthreads();
    }

    // C/D layout: VGPR v, lane -> M = v + 8*half, N = r
    #pragma unroll
    for (int v = 0; v < 8; ++v) {
        int row = rowBase + v + 8 * half;
        atomicAdd(&alphas[row * ACOLS + r],       acc0[v]);
        atomicAdd(&alphas[row * ACOLS + 16 + r],  acc1[v]);
    }
}

// ---------------------------------------------------------------------------
// Kernel 2: one block per batch element; build masks from 26 alphas.
__global__ void mask_kernel(const float* __restrict__ alphas,
                            float* __restrict__ out) {
    __shared__ float sA[26];      // raw alphas
    __shared__ float sAct[26];    // activations
    __shared__ float sC[4];       // rot polynomial coefficients
    __shared__ float tmx[100], tmy[100], smx[100], smy[100];

    const int t = threadIdx.x;
    const int b = blockIdx.x;

    if (t < 26) sA[t] = alphas[b * ACOLS + t];
    __syncthreads();

    if (t == 0) {
        for (int k = 0; k < 13; ++k) sAct[k] = 1.0f / (1.0f + expf(-sA[k]));
        sAct[21] = 1.0f / (1.0f + expf(-sA[21]));
        const int starts[3] = {13, 17, 22};   // sc_ax, sc_ay, agg softmaxes
        for (int g = 0; g < 3; ++g) {
            int s0 = starts[g];
            float m = sA[s0];
            for (int k = 1; k < 4; ++k) m = fmaxf(m, sA[s0 + k]);
            float e[4], sum = 0.0f;
            for (int k = 0; k < 4; ++k) { e[k] = expf(sA[s0 + k] - m); sum += e[k]; }
            for (int k = 0; k < 4; ++k) sAct[s0 + k] = e[k] / sum;
        }
        // rot_mask = prod_i (a_i*ROT + (1-a_i)*I) = c0 I + c1 R + c2 R^2 + c3 R^3
        float a0 = sAct[8], a1 = sAct[9], a2 = sAct[10];
        float q0 = 1.0f - a0, q1 = 1.0f - a1, q2 = 1.0f - a2;
        sC[0] = q0 * q1 * q2;
        sC[1] = a0 * q1 * q2 + q0 * a1 * q2 + q0 * q1 * a2;
        sC[2] = a0 * a1 * q2 + a0 * q1 * a2 + q0 * a1 * a2;
        sC[3] = a0 * a1 * a2;
    }
    __syncthreads();

    const int i = t / 10, j = t % 10;        // valid for t < 100
    if (t < 100) { float e = (i == j) ? 1.0f : 0.0f; tmx[t] = e; tmy[t] = e; }
    __syncthreads();

    // TRANSLATE[idx] shifts by -2^(3-idx): row gather with shift {2,6,8,9} mod 10
    #pragma unroll
    for (int step = 0; step < 4; ++step) {
        const int shiftArr[4] = {2, 6, 8, 9};
        float nvx = 0.0f, nvy = 0.0f;
        if (t < 100) {
            int sr = (i + shiftArr[step]) % 10;
            float ax = sAct[step], ay = sAct[4 + step];
            nvx = ax * tmx[sr * 10 + j] + (1.0f - ax) * tmx[t];
            nvy = ay * tmy[sr * 10 + j] + (1.0f - ay) * tmy[t];
        }
        __syncthreads();
        if (t < 100) { tmx[t] = nvx; tmy[t] = nvy; }
        __syncthreads();
    }

    // SCALE_h[i, i/h] = 1  ->  smx[i][j] = sum_h w_h * (j == i/h)
    if (t < 100) {
        float vx = 0.0f, vy = 0.0f;
        #pragma unroll
        for (int h = 1; h <= 4; ++h) {
            float hit = (j == i / h) ? 1.0f : 0.0f;
            vx += sAct[12 + h] * hit;    // sc_ax at sAct[13..16]
            vy += sAct[16 + h] * hit;    // sc_ay at sAct[17..20]
        }
        smx[t] = vx; smy[t] = vy;
    }
    __syncthreads();

    const float refax = sAct[11], refay = sAct[12], scat = sAct[21];
    const float g0 = sAct[22], g1 = sAct[23], g2 = sAct[24], g3 = sAct[25];
    const float c0 = sC[0], c1 = sC[1], c2 = sC[2], c3 = sC[3];

    float* outb = out + (size_t)b * 10000;
    for (int l = t; l < 10000; l += blockDim.x) {
        int oi = l / 100, oj = l % 100;
        int i1 = oi / 10, i2 = oi % 10, j1 = oj / 10, j2 = oj % 10;

        float trans = tmx[i1 * 10 + j1] * tmy[i2 * 10 + j2];

        float rmx = refax * ((j1 == 9 - i1) ? 1.0f : 0.0f)
                  + (1.0f - refax) * ((j1 == i1) ? 1.0f : 0.0f);
        float rmy = refay * ((j2 == 9 - i2) ? 1.0f : 0.0f)
                  + (1.0f - refay) * ((j2 == i2) ? 1.0f : 0.0f);
        float refv = rmx * rmy;

        // ROT permutation p(i) = (10*i + 9 - i/10) mod 100
        int p1 = (10 * oi + 9 - i1) % 100;
        int p2 = (10 * p1 + 9 - p1 / 10) % 100;
        int p3 = (10 * p2 + 9 - p2 / 10) % 100;
        float rotv = c0 * (float)(oj == oi) + c1 * (float)(oj == p1)
                   + c2 * (float)(oj == p2) + c3 * (float)(oj == p3);

        float smij = smx[i1 * 10 + j1] * smy[i2 * 10 + j2];
        float smji = smx[j1 * 10 + i1] * smy[j2 * 10 + i2];
        float scalev = scat * smji + (1.0f - scat) * smij;

        outb[l] = g0 * trans + g1 * refv + g2 * rotv + g3 * scalev;
    }
}

// ---------------------------------------------------------------------------
extern "C" void kernel_launch(void* const* d_in, const int* in_sizes, int n_in,
                              void* d_out, int out_size, void* d_ws, size_t ws_size,
                              hipStream_t stream) {
    const float* x  = (const float*)d_in[0];
    const float* W  = (const float*)d_in[1];
    const float* bb = (const float*)d_in[2];
    float* out    = (float*)d_out;
    float* alphas = (float*)d_ws;            // BATCH*ACOLS floats = 256 KB

    init_alphas_kernel<<<(BATCH * ACOLS + 255) / 256, 256, 0, stream>>>(bb, alphas);
    gemm_alphas_kernel<<<dim3(BATCH / 16, KSPLIT), 128, 0, stream>>>(x, W, alphas);
    mask_kernel<<<BATCH, 128, 0, stream>>>(alphas, out);
}